// GNNModel_17008070493041
// MI455X (gfx1250) — compile-verified
//
#include <hip/hip_runtime.h>
#include <hip/hip_bf16.h>

typedef float v2f __attribute__((ext_vector_type(2)));
typedef float v8f __attribute__((ext_vector_type(8)));

// ---------------------------------------------------------------------------
// Degree / normalization
// ---------------------------------------------------------------------------
__global__ void init_deg_kernel(float* __restrict__ deg, int n) {
    int i = blockIdx.x * blockDim.x + threadIdx.x;
    if (i < n) deg[i] = 1.0f;                      // self-loop contributes 1
}

__global__ void count_deg_kernel(const long long* __restrict__ dst,
                                 float* __restrict__ deg, long long e_cnt) {
    long long e = (long long)blockIdx.x * blockDim.x + threadIdx.x;
    if (e < e_cnt) atomicAdd(&deg[dst[e]], 1.0f);
}

__global__ void finish_dinv_kernel(float* __restrict__ deg, int n) {
    int i = blockIdx.x * blockDim.x + threadIdx.x;
    if (i < n) deg[i] = rsqrtf(deg[i]);            // deg >= 1 always
}

// ---------------------------------------------------------------------------
// GEMM1: x1 = x @ W1   [N,128] x [128,32] -> [N,32]
// One wave computes a 16x32 tile via v_wmma_f32_16x16x4_f32 (two N-tiles).
// A 16x4 f32 frag: lanes 0-15 hold K=(k0,k0+1), lanes 16-31 K=(k0+2,k0+3).
// B 4x16 frag: lane%16 = column, lane half selects K pair.
// W is staged in LDS pre-packed as K-pair float2s so each B fragment is a
// single contiguous ds_load_b64 into an aligned VGPR pair (no repack movs).
// C/D: VGPR r -> row (r + 8*(lane>=16)), col lane%16.
// ---------------------------------------------------------------------------
__global__ void __launch_bounds__(256) gemm1_wmma_kernel(
        const float* __restrict__ x, const float* __restrict__ W1,
        float* __restrict__ x1, int n) {
    __shared__ v2f sW[64 * 32];                    // sW[p*32+n] = {W1[2p][n], W1[2p+1][n]}
    for (int i = threadIdx.x; i < 64 * 32; i += 256) {
        int p = i >> 5, col = i & 31;
        v2f w;
        w.x = W1[(2 * p)     * 32 + col];
        w.y = W1[(2 * p + 1) * 32 + col];
        sW[i] = w;
    }
    __syncthreads();

    const int wave = threadIdx.x >> 5;
    const int lane = threadIdx.x & 31;
    const int tile = blockIdx.x * 8 + wave;        // 16-row tile id
    if (tile * 16 >= n) return;                    // wave-uniform

    const int row0 = tile * 16;
    const int m    = lane & 15;
    const int koff = (lane >> 4) << 1;             // 0 or 2
    const float* xrow = x + (size_t)(row0 + m) * 128;

    v8f c0 = {}; v8f c1 = {};
    for (int k0 = 0; k0 < 128; k0 += 4) {
        v2f a  = *(const v2f*)(xrow + k0 + koff);  // 8B aligned, single b64 load
        int p  = (k0 + koff) >> 1;                 // K-pair index
        v2f b0 = sW[p * 32 + m];
        v2f b1 = sW[p * 32 + 16 + m];
        c0 = __builtin_amdgcn_wmma_f32_16x16x4_f32(false, a, false, b0,
                                                   (short)0, c0, false, false);
        c1 = __builtin_amdgcn_wmma_f32_16x16x4_f32(false, a, false, b1,
                                                   (short)0, c1, false, false);
    }

    const int mstore = (lane >> 4) * 8;            // 0 or 8
    float* o = x1 + (size_t)(row0 + mstore) * 32 + (lane & 15);
#pragma unroll
    for (int r = 0; r < 8; ++r) {
        o[(size_t)r * 32]      = c0[r];
        o[(size_t)r * 32 + 16] = c1[r];
    }
}

// ---------------------------------------------------------------------------
// Self-loop term doubles as aggregation-buffer init: agg[i] = x1[i]*dinv[i]^2
// ---------------------------------------------------------------------------
__global__ void selfloop32_kernel(const float* __restrict__ x1,
                                  const float* __restrict__ dinv,
                                  float* __restrict__ agg, int n) {
    long long g = (long long)blockIdx.x * blockDim.x + threadIdx.x;
    long long i = g >> 3;
    if (i >= n) return;
    int q = (int)(g & 7);
    float w = dinv[i]; w *= w;
    float4 v = *(const float4*)(x1 + i * 32 + q * 4);
    float4 r = make_float4(v.x * w, v.y * w, v.z * w, v.w * w);
    *(float4*)(agg + i * 32 + q * 4) = r;
}

__global__ void selfloop16_kernel(const float* __restrict__ x2,
                                  const float* __restrict__ dinv,
                                  float* __restrict__ agg, int n) {
    long long g = (long long)blockIdx.x * blockDim.x + threadIdx.x;
    long long i = g >> 2;
    if (i >= n) return;
    int q = (int)(g & 3);
    float w = dinv[i]; w *= w;
    float4 v = *(const float4*)(x2 + i * 16 + q * 4);
    float4 r = make_float4(v.x * w, v.y * w, v.z * w, v.w * w);
    *(float4*)(agg + i * 16 + q * 4) = r;
}

// ---------------------------------------------------------------------------
// Edge scatter: 8 threads/edge (32-wide feats) or 4 threads/edge (16-wide).
// Consecutive lanes of an edge read a contiguous float4 slice of the source
// row -> coalesced 128B gather; atomics hit L2 (agg fits in 192MB L2).
// ---------------------------------------------------------------------------
__global__ void __launch_bounds__(256) scatter32_kernel(
        const long long* __restrict__ src, const long long* __restrict__ dst,
        const float* __restrict__ dinv, const float* __restrict__ x1,
        float* __restrict__ agg, long long e_cnt) {
    long long g = (long long)blockIdx.x * blockDim.x + threadIdx.x;
    long long e = g >> 3;
    if (e >= e_cnt) return;
    int q = (int)(g & 7);
    long long s = src[e], d = dst[e];
    float nrm = dinv[s] * dinv[d];
    float4 v = *(const float4*)(x1 + s * 32 + q * 4);
    float* o = agg + d * 32 + q * 4;
    atomicAdd(o + 0, v.x * nrm);
    atomicAdd(o + 1, v.y * nrm);
    atomicAdd(o + 2, v.z * nrm);
    atomicAdd(o + 3, v.w * nrm);
}

__global__ void __launch_bounds__(256) scatter16_kernel(
        const long long* __restrict__ src, const long long* __restrict__ dst,
        const float* __restrict__ dinv, const float* __restrict__ x2,
        float* __restrict__ agg, long long e_cnt) {
    long long g = (long long)blockIdx.x * blockDim.x + threadIdx.x;
    long long e = g >> 2;
    if (e >= e_cnt) return;
    int q = (int)(g & 3);
    long long s = src[e], d = dst[e];
    float nrm = dinv[s] * dinv[d];
    float4 v = *(const float4*)(x2 + s * 16 + q * 4);
    float* o = agg + d * 16 + q * 4;
    atomicAdd(o + 0, v.x * nrm);
    atomicAdd(o + 1, v.y * nrm);
    atomicAdd(o + 2, v.z * nrm);
    atomicAdd(o + 3, v.w * nrm);
}

// ---------------------------------------------------------------------------
// GEMM2: x2 = relu(agg1 + b1) @ W2   [N,32] x [32,16] -> [N,16]
// bias+ReLU fused into the A-fragment load; W2 and b1 staged as K-pair v2f.
// ---------------------------------------------------------------------------
__global__ void __launch_bounds__(256) gemm2_wmma_kernel(
        const float* __restrict__ agg1, const float* __restrict__ W2,
        const float* __restrict__ b1, float* __restrict__ x2, int n) {
    __shared__ v2f sW[16 * 16];                    // sW[p*16+n] = {W2[2p][n], W2[2p+1][n]}
    __shared__ v2f sb[16];                         // sb[p] = {b1[2p], b1[2p+1]}
    if (threadIdx.x < 16 * 16) {
        int p = threadIdx.x >> 4, col = threadIdx.x & 15;
        v2f w;
        w.x = W2[(2 * p)     * 16 + col];
        w.y = W2[(2 * p + 1) * 16 + col];
        sW[threadIdx.x] = w;
    }
    if (threadIdx.x < 16) {
        v2f t; t.x = b1[2 * threadIdx.x]; t.y = b1[2 * threadIdx.x + 1];
        sb[threadIdx.x] = t;
    }
    __syncthreads();

    const int wave = threadIdx.x >> 5;
    const int lane = threadIdx.x & 31;
    const int tile = blockIdx.x * 8 + wave;
    if (tile * 16 >= n) return;

    const int row0 = tile * 16;
    const int m    = lane & 15;
    const int koff = (lane >> 4) << 1;
    const float* arow = agg1 + (size_t)(row0 + m) * 32;

    v8f c = {};
    for (int k0 = 0; k0 < 32; k0 += 4) {
        int p  = (k0 + koff) >> 1;
        v2f av = *(const v2f*)(arow + k0 + koff);
        v2f bv = sb[p];
        v2f a;
        a.x = fmaxf(av.x + bv.x, 0.0f);
        a.y = fmaxf(av.y + bv.y, 0.0f);
        v2f b = sW[p * 16 + m];
        c = __builtin_amdgcn_wmma_f32_16x16x4_f32(false, a, false, b,
                                                  (short)0, c, false, false);
    }

    const int mstore = (lane >> 4) * 8;
    float* o = x2 + (size_t)(row0 + mstore) * 16 + (lane & 15);
#pragma unroll
    for (int r = 0; r < 8; ++r) o[(size_t)r * 16] = c[r];
}

// ---------------------------------------------------------------------------
// Head: h2 = relu(agg2 + b2); logits = h2 @ Wfc + bfc; log_softmax(2 cls)
// ---------------------------------------------------------------------------
__global__ void head_kernel(const float* __restrict__ agg2,
                            const float* __restrict__ b2,
                            const float* __restrict__ Wfc,
                            const float* __restrict__ bfc,
                            float* __restrict__ out, int n) {
    int i = blockIdx.x * blockDim.x + threadIdx.x;
    if (i >= n) return;
    const float* r = agg2 + (size_t)i * 16;
    float l0 = bfc[0], l1 = bfc[1];
#pragma unroll
    for (int k = 0; k < 16; ++k) {
        float h = fmaxf(r[k] + b2[k], 0.0f);
        l0 += h * Wfc[k * 2 + 0];
        l1 += h * Wfc[k * 2 + 1];
    }
    float mx  = fmaxf(l0, l1);
    float lse = mx + logf(expf(l0 - mx) + expf(l1 - mx));
    out[(size_t)i * 2 + 0] = l0 - lse;
    out[(size_t)i * 2 + 1] = l1 - lse;
}

// ---------------------------------------------------------------------------
extern "C" void kernel_launch(void* const* d_in, const int* in_sizes, int n_in,
                              void* d_out, int out_size, void* d_ws, size_t ws_size,
                              hipStream_t stream) {
    const float*     x    = (const float*)d_in[0];
    const long long* edge = (const long long*)d_in[1];
    const float*     W1   = (const float*)d_in[2];
    const float*     b1   = (const float*)d_in[3];
    const float*     W2   = (const float*)d_in[4];
    const float*     b2   = (const float*)d_in[5];
    const float*     Wfc  = (const float*)d_in[6];
    const float*     bfc  = (const float*)d_in[7];
    float* out = (float*)d_out;

    const int       N = in_sizes[0] / 128;          // 500000 (multiple of 16)
    const long long E = (long long)in_sizes[1] / 2; // 8000000
    const long long* src = edge;
    const long long* dst = edge + E;

    float* ws   = (float*)d_ws;
    float* dinv = ws;                                // N
    float* x1   = ws + N;                            // N*32
    float* agg1 = x1 + (size_t)N * 32;               // N*32
    float* x2   = x1;                                // reuse (N*16)
    float* agg2 = x1 + (size_t)N * 16;               // reuse (N*16)

    const int B = 256;
    // degrees + rsqrt normalization
    init_deg_kernel   <<<(N + B - 1) / B, B, 0, stream>>>(dinv, N);
    count_deg_kernel  <<<(int)((E + B - 1) / B), B, 0, stream>>>(dst, dinv, E);
    finish_dinv_kernel<<<(N + B - 1) / B, B, 0, stream>>>(dinv, N);

    // layer 1
    int tiles = N / 16;
    gemm1_wmma_kernel<<<(tiles + 7) / 8, B, 0, stream>>>(x, W1, x1, N);
    selfloop32_kernel<<<(int)(((long long)N * 8 + B - 1) / B), B, 0, stream>>>(x1, dinv, agg1, N);
    scatter32_kernel <<<(int)((E * 8 + B - 1) / B), B, 0, stream>>>(src, dst, dinv, x1, agg1, E);

    // layer 2 (relu+bias fused into GEMM2 A-load)
    gemm2_wmma_kernel<<<(tiles + 7) / 8, B, 0, stream>>>(agg1, W2, b1, x2, N);
    selfloop16_kernel<<<(int)(((long long)N * 4 + B - 1) / B), B, 0, stream>>>(x2, dinv, agg2, N);
    scatter16_kernel <<<(int)((E * 4 + B - 1) / B), B, 0, stream>>>(src, dst, dinv, x2, agg2, E);

    // head: relu + FC + log_softmax
    head_kernel<<<(N + B - 1) / B, B, 0, stream>>>(agg2, b2, Wfc, bfc, out, N);
}